// PositionalEncoding_16252156248517
// MI455X (gfx1250) — compile-verified
//
#include <hip/hip_runtime.h>
#include <hip/hip_bf16.h>

typedef __attribute__((ext_vector_type(4))) float        v4f;
typedef __attribute__((ext_vector_type(4))) unsigned int u32x4;
typedef __attribute__((ext_vector_type(8))) int          i32x8;
typedef __attribute__((ext_vector_type(4))) int          i32x4;

#define PE_DIM 1024
#define PE_B   8
#define SPLIT_TOK 49

// TDM path only on the device pass for gfx1250 where the builtin exists.
#if defined(__gfx1250__) && __has_builtin(__builtin_amdgcn_tensor_load_to_lds)
#define USE_TDM 1
#if __has_include(<hip/amd_detail/amd_gfx1250_TDM.h>)
#define TDM_6ARG 1   // amdgpu-toolchain (clang-23): 6-arg builtin
#endif
#endif

__global__ void __launch_bounds__(256)
pe_fused_kernel(const float* __restrict__ emb,
                const int*   __restrict__ src_org,
                const float* __restrict__ pe,
                float* __restrict__ out,        // [S, B, DIM]
                float* __restrict__ doc_out,    // [S, B, 1]
                int S) {
    const int tid = threadIdx.x;

    // ---------------- doc_emb scan block (last block) ----------------
    if ((int)blockIdx.x == S) {
        const int wave = tid >> 5;   // 8 waves <-> 8 batch columns
        const int lane = tid & 31;
        if (wave < PE_B) {
            const int b = wave;
            float running = 0.0f;
            for (int base = 0; base < S; base += 32) {
                const int t = base + lane;
                bool m = false;
                if (t < S) m = (src_org[(size_t)t * PE_B + b] == SPLIT_TOK);
                const unsigned bal = (unsigned)__ballot(m);          // wave32
                const unsigned le_mask = 0xFFFFFFFFu >> (31 - lane); // lanes <= me
                const float cum = running + (float)__popc(bal & le_mask);
                if (t < S) {
                    bool m_next = (t + 1 < S) &&
                                  (src_org[(size_t)(t + 1) * PE_B + b] == SPLIT_TOK);
                    doc_out[(size_t)t * PE_B + b] = m_next ? 0.0f : cum;
                }
                running += (float)__popc(bal);   // wave-uniform update
            }
        }
        return;
    }

    // ---------------- main elementwise block: one block per s ----------------
    const int s = blockIdx.x;
    __shared__ __align__(16) float pe_lds[PE_DIM];

    // Issue all 8 streaming emb loads FIRST so their latency overlaps the
    // TDM transfer + barrier below (compiler can't hoist vmem across barrier).
    const int d0 = tid * 4;                     // 256 threads * 4 floats = 1024
    const size_t row_base = (size_t)s * PE_B * PE_DIM + d0;
    v4f e[PE_B];
#pragma unroll
    for (int b = 0; b < PE_B; ++b)
        e[b] = __builtin_nontemporal_load((const v4f*)(emb + row_base + (size_t)b * PE_DIM));

#if defined(USE_TDM)
    if ((tid >> 5) == 0) {  // wave 0 only: TDM is per-wave, EXEC-agnostic
        const unsigned long long gaddr =
            (unsigned long long)(uintptr_t)(pe + (size_t)s * PE_DIM);
        const unsigned lds_addr = (unsigned)(uintptr_t)&pe_lds[0]; // low 32b of flat = LDS byte addr

        // D# group 0 (4 dwords): count=1 | lds_addr | global_addr[56:0] | type=2
        u32x4 g0;
        g0.x = 1u;                                        // count=1, user mode
        g0.y = lds_addr;                                  // bits 63:32
        g0.z = (unsigned)(gaddr & 0xFFFFFFFFull);         // bits 95:64
        g0.w = (unsigned)(gaddr >> 32) | (2u << 30);      // addr[56:32] | type=2

        // D# group 1 (8 dwords): data_size=4B, tensor 1024x1, tile 1024x1, stride0=1024
        i32x8 g1;
        g1[0] = (int)(2u << 16);                 // data_size=2 (4 bytes), wg_mask=0
        g1[1] = (int)((PE_DIM & 0xFFFF) << 16);  // tensor_dim0[15:0] @ bits 63:48
        g1[2] = (int)((PE_DIM >> 16) | (1u << 16)); // tensor_dim0[31:16] | tensor_dim1[15:0]=1
        g1[3] = (int)((unsigned)PE_DIM << 16);   // tensor_dim1[31:16]=0 | tile_dim0=1024
        g1[4] = 1;                               // tile_dim1=1 | tile_dim2=0
        g1[5] = PE_DIM;                          // tensor_dim0_stride[31:0]
        g1[6] = 0;                               // stride0[47:32] | stride1[15:0]
        g1[7] = 0;                               // stride1[47:16]

        i32x4 gz = {0, 0, 0, 0};
#if defined(TDM_6ARG)
        i32x8 gz8 = {0, 0, 0, 0, 0, 0, 0, 0};
        __builtin_amdgcn_tensor_load_to_lds(g0, g1, gz, gz, gz8, 0);
#else
        __builtin_amdgcn_tensor_load_to_lds(g0, g1, gz, gz, 0);
#endif
        __builtin_amdgcn_s_wait_tensorcnt(0);
    }
#else
    // Fallback (host pass / no TDM builtin): plain coalesced LDS fill.
    {
        *(v4f*)&pe_lds[d0] = *(const v4f*)&pe[(size_t)s * PE_DIM + d0];
    }
#endif
    __syncthreads();

    const v4f p = *(const v4f*)&pe_lds[d0];     // ds_load_b128, reused 8x

#pragma unroll
    for (int b = 0; b < PE_B; ++b) {
        v4f o;
        o.x = __builtin_fmaf(e[b].x, 32.0f, p.x);  // sqrt(1024) == 32 exactly
        o.y = __builtin_fmaf(e[b].y, 32.0f, p.y);
        o.z = __builtin_fmaf(e[b].z, 32.0f, p.z);
        o.w = __builtin_fmaf(e[b].w, 32.0f, p.w);
        __builtin_nontemporal_store(o, (v4f*)(out + row_base + (size_t)b * PE_DIM)); // streaming write
    }
}

extern "C" void kernel_launch(void* const* d_in, const int* in_sizes, int n_in,
                              void* d_out, int out_size, void* d_ws, size_t ws_size,
                              hipStream_t stream) {
    const float* emb     = (const float*)d_in[0];
    const int*   src_org = (const int*)d_in[1];
    const float* pe      = (const float*)d_in[2];

    const int S = in_sizes[0] / (PE_B * PE_DIM);      // 4096
    float* out     = (float*)d_out;                   // [S,B,DIM]
    float* doc_out = (float*)d_out + (size_t)S * PE_B * PE_DIM; // [S,B,1]

    // Blocks 0..S-1: one pe-row each. Block S: doc_emb wave-scan.
    pe_fused_kernel<<<S + 1, 256, 0, stream>>>(emb, src_org, pe, out, doc_out, S);
}